// CMAlign_25786983645449
// MI455X (gfx1250) — compile-verified
//
#include <hip/hip_runtime.h>
#include <hip/hip_bf16.h>

#define NTOT 64
#define CCH  2048
#define HW   162
#define SP   176     // spatial padded to 11 tiles of 16
#define TILES 11
#define BLK2 352     // 11 waves (wave32)
#define PANEL (32 * SP)

typedef __attribute__((ext_vector_type(2))) float v2f;
typedef __attribute__((ext_vector_type(8))) float v8f;
typedef __attribute__((ext_vector_type(4))) unsigned int u32x4;
typedef __attribute__((ext_vector_type(8))) int i32x8;
typedef __attribute__((ext_vector_type(4))) int i32x4;

#define INF __builtin_inff()

// workspace layout (in floats)
#define WS_INV  0
#define WS_MASK (WS_INV  + NTOT*SP)
#define WS_MW   (WS_MASK + NTOT*SP)
#define WS_DSQ  (WS_MW   + NTOT*SP)
#define WS_CM   (WS_DSQ  + 2*NTOT*SP)
#define WS_TRIP (WS_CM   + NTOT*SP)
#define WS_MPT  (WS_TRIP + NTOT*SP)

__device__ __forceinline__ const float* fbase(const float* fv, const float* ft, int n) {
  return (n < 32) ? (fv + (size_t)n * CCH * HW) : (ft + (size_t)(n - 32) * CCH * HW);
}

__device__ __forceinline__ unsigned int lds_off(const void* p) {
  // generic pointer to LDS: low 32 bits are the LDS byte offset
  return (unsigned int)(uintptr_t)p;
}

// TDM: 2D tile (tile_d0 x tile_d1) of 4-byte elements, row stride stride0 (elements),
// zero-fill for x >= tensor_d0. Issued per-wave (EXEC ignored); call from one wave only.
__device__ __forceinline__ void tdm_load_2d(unsigned int dst_lds, const void* gsrc,
                                            unsigned int tensor_d0, unsigned int tensor_d1,
                                            unsigned int tile_d0, unsigned int tile_d1,
                                            unsigned int stride0) {
  const unsigned long long ga = (unsigned long long)(uintptr_t)gsrc;
  u32x4 g0;
  g0.x = 1u;                                                  // count=1 (valid), user D#
  g0.y = dst_lds;                                             // lds_addr (bytes)
  g0.z = (unsigned int)ga;                                    // global_addr[31:0]
  g0.w = (unsigned int)((ga >> 32) & 0x1FFFFFFu) | (2u << 30); // addr[56:32] | type=2
  i32x8 g1;
  g1[0] = (int)(2u << 16);                                    // data_size = 4 bytes
  g1[1] = (int)((tensor_d0 & 0xFFFFu) << 16);                 // tensor_dim0[15:0]
  g1[2] = (int)(((tensor_d0 >> 16) & 0xFFFFu) | ((tensor_d1 & 0xFFFFu) << 16));
  g1[3] = (int)(((tensor_d1 >> 16) & 0xFFFFu) | ((tile_d0 & 0xFFFFu) << 16));
  g1[4] = (int)(tile_d1 & 0xFFFFu);                           // tile_dim1 (tile_dim2 = 0)
  g1[5] = (int)stride0;                                       // tensor_dim0_stride[31:0]
  g1[6] = 0;
  g1[7] = 0;
  const i32x4 z4 = {0, 0, 0, 0};
#if __clang_major__ >= 23
  const i32x8 z8 = {0, 0, 0, 0, 0, 0, 0, 0};
  __builtin_amdgcn_tensor_load_to_lds(g0, g1, z4, z4, z8, 0);
#else
  __builtin_amdgcn_tensor_load_to_lds(g0, g1, z4, z4, 0);
#endif
}

// ---------------- Kernel 1: per-pixel channel norms -> inv-norm + mask ----------------
__global__ __launch_bounds__(192) void k_norm(const float* __restrict__ fv,
                                              const float* __restrict__ ft,
                                              float* __restrict__ ws) {
  const int n = blockIdx.x;
  const int t = threadIdx.x;
  const float* f = fbase(fv, ft, n);
  float acc = 0.f;
  if (t < HW) {
    const float* p = f + t;
#pragma unroll 8
    for (int c = 0; c < CCH; ++c) {
      float v = p[(size_t)c * HW];
      acc += v * v;
    }
  }
  const float nrm = sqrtf(acc);
  __shared__ float smin[192], smax[192];
  smin[t] = (t < HW) ? nrm : INF;
  smax[t] = (t < HW) ? nrm : -INF;
  __syncthreads();
  if (t < 96) { smin[t] = fminf(smin[t], smin[t + 96]); smax[t] = fmaxf(smax[t], smax[t + 96]); }
  __syncthreads();
  for (int s = 48; s > 0; s >>= 1) {
    if (t < s) { smin[t] = fminf(smin[t], smin[t + s]); smax[t] = fmaxf(smax[t], smax[t + s]); }
    __syncthreads();
  }
  const float mn = smin[0], mx = smax[0];
  if (t < SP) {
    float iv = 0.f, mk = 0.f;
    if (t < HW) {
      iv = 1.f / fmaxf(nrm, 1e-12f);
      mk = (nrm - mn) / ((mx - mn) + 1e-12f);
    }
    ws[WS_INV  + n * SP + t] = iv;
    ws[WS_MASK + n * SP + t] = mk;
  }
}

// ---------------- Kernel 2: sim GEMM (WMMA f32, TDM double-buffered) + softmax ----------------
__global__ __launch_bounds__(BLK2) void k_sim(const float* __restrict__ fv,
                                              const float* __restrict__ ft,
                                              const int* __restrict__ pidx,
                                              const int* __restrict__ nidx,
                                              float* __restrict__ ws) {
  const int branch = blockIdx.x & 1;
  const int n = blockIdx.x >> 1;
  const int p = branch ? nidx[n] : pidx[n];
  const float* fN = fbase(fv, ft, n);
  const float* fP = fbase(fv, ft, p);
  const int tid = threadIdx.x;
  const int w = tid >> 5, lane = tid & 31, ll = lane & 15, hi = lane >> 4;

  extern __shared__ float lds[];
  float* la   = lds;                  // 2 * PANEL : raw f[p] chunks [k][t]
  float* lb   = la + 2 * PANEL;       // 2 * PANEL : raw f[n] chunks [k][s]
  float* invp = lb + 2 * PANEL;       // SP
  float* invn = invp + SP;            // SP
  float* mskn = invn + SP;            // SP

  if (tid < SP) {
    invp[tid] = ws[WS_INV  + p * SP + tid];
    invn[tid] = ws[WS_INV  + n * SP + tid];
    mskn[tid] = ws[WS_MASK + n * SP + tid];
  }

  // prologue: async-DMA chunk 0 into buffer 0 (wave 0 only; EXEC ignored by TDM)
  if (w == 0) {
    tdm_load_2d(lds_off(la), fP, HW, CCH, SP, 32, HW);
    tdm_load_2d(lds_off(lb), fN, HW, CCH, SP, 32, HW);
  }

  v8f acc[TILES] = {};

  for (int ch = 0; ch < CCH / 32; ++ch) {
    const int cur = ch & 1;
    if (w == 0) __builtin_amdgcn_s_wait_tensorcnt(0);
    __syncthreads();                         // publish buffer 'cur'; prior compute done
    if (w == 0 && ch + 1 < CCH / 32) {       // prefetch next chunk into the other buffer
      const float* srcP = fP + (size_t)(ch + 1) * 32 * HW;
      const float* srcN = fN + (size_t)(ch + 1) * 32 * HW;
      tdm_load_2d(lds_off(la + (cur ^ 1) * PANEL), srcP, HW, CCH, SP, 32, HW);
      tdm_load_2d(lds_off(lb + (cur ^ 1) * PANEL), srcN, HW, CCH, SP, 32, HW);
    }
    const float* A = la + cur * PANEL;
    const float* B = lb + cur * PANEL;
#pragma unroll
    for (int k0 = 0; k0 < 32; k0 += 4) {
      v2f a;
      a.x = A[(k0 + 2 * hi)     * SP + w * 16 + ll];
      a.y = A[(k0 + 2 * hi + 1) * SP + w * 16 + ll];
#pragma unroll
      for (int nt = 0; nt < TILES; ++nt) {
        v2f b;
        b.x = B[(k0 + 2 * hi)     * SP + nt * 16 + ll];
        b.y = B[(k0 + 2 * hi + 1) * SP + nt * 16 + ll];
        acc[nt] = __builtin_amdgcn_wmma_f32_16x16x4_f32(
            false, a, false, b, (short)0, acc[nt], false, false);
      }
    }
  }

  // epilogue: scale raw dot products by inv_p[t]*inv_n[s], softmax over s, mask_warp.
  float ivn_l[TILES], msk_l[TILES];
#pragma unroll
  for (int nt = 0; nt < TILES; ++nt) {
    ivn_l[nt] = invn[nt * 16 + ll];
    msk_l[nt] = mskn[nt * 16 + ll];
  }
  float* mpT = ws + WS_MPT + (size_t)(branch * NTOT + n) * SP * SP;
  for (int r = 0; r < 8; ++r) {
    const int t = w * 16 + r + 8 * hi;
    const float ivp_t = invp[t];
    float vv[TILES];
    float rmax = -INF;
#pragma unroll
    for (int nt = 0; nt < TILES; ++nt) {
      float v = ((float*)&acc[nt])[r] * ivp_t * ivn_l[nt];
      const bool sv = (nt < 10) | (ll < 2);          // s = nt*16+ll < 162 ?
      v = sv ? v : -INF;
      vv[nt] = v;
      rmax = fmaxf(rmax, v);
    }
#pragma unroll
    for (int m = 1; m < 16; m <<= 1) rmax = fmaxf(rmax, __shfl_xor(rmax, m, 16));
    float rsum = 0.f;
#pragma unroll
    for (int nt = 0; nt < TILES; ++nt) {
      const bool sv = (nt < 10) | (ll < 2);
      const float e = sv ? __expf(50.f * (vv[nt] - rmax)) : 0.f;
      vv[nt] = e; rsum += e;
    }
#pragma unroll
    for (int m = 1; m < 16; m <<= 1) rsum += __shfl_xor(rsum, m, 16);
    const float rinv = 1.f / rsum;
    float mw = 0.f;
#pragma unroll
    for (int nt = 0; nt < TILES; ++nt) {
      const int s = nt * 16 + ll;
      const float mp = vv[nt] * rinv;
      mpT[(size_t)s * SP + t] = mp;                  // full padded tile (pad s rows are 0)
      mw += mp * msk_l[nt];
    }
    if (branch == 0) {
#pragma unroll
      for (int m = 1; m < 16; m <<= 1) mw += __shfl_xor(mw, m, 16);
      if (ll == 0 && t < HW) ws[WS_MW + n * SP + t] = mw;
    }
  }
}

// ---------------- Kernel 3: warped GEMM (WMMA f32, TDM for mp^T) + recon + distances ----------------
__global__ __launch_bounds__(BLK2) void k_warp(const float* __restrict__ fv,
                                               const float* __restrict__ ft,
                                               float* __restrict__ ws,
                                               float* __restrict__ out) {
  const int branch = blockIdx.x & 1;
  const int n = blockIdx.x >> 1;
  const float* fN = fbase(fv, ft, n);
  const int tid = threadIdx.x;
  const int w = tid >> 5, lane = tid & 31, ll = lane & 15, hi = lane >> 4;

  extern __shared__ float lds[];
  float* Bsh = lds;                      // SP*SP   : mp^T [s][t]
  float* Ash = Bsh + SP * SP;            // 16*177  : f tile [c_local][s], padded stride
  float* msk = Ash + 16 * 177;           // SP

  const float* mpT = ws + WS_MPT + (size_t)(branch * NTOT + n) * SP * SP;
  if (w == 0) {                          // single 124KB DMA: fully-initialized padded tile
    tdm_load_2d(lds_off(Bsh), mpT, SP * SP, 1, SP * SP, 1, SP * SP);
  }
  if (tid < SP) msk[tid] = (tid < HW) ? ws[WS_MASK + n * SP + tid] : 0.f;
  if (w == 0) __builtin_amdgcn_s_wait_tensorcnt(0);

  const int nt16 = w * 16;
  const int t = nt16 + ll;
  const bool tv = t < HW;
  float dacc = 0.f;

  for (int m = 0; m < CCH / 16; ++m) {
    __syncthreads();
    for (int e = tid; e < 16 * SP; e += BLK2) {
      const int k = e / SP, s = e - k * SP;
      Ash[k * 177 + s] = (s < HW) ? fN[(size_t)(m * 16 + k) * HW + s] : 0.f;
    }
    __syncthreads();

    v8f a0 = {}, a1 = {};
#pragma unroll
    for (int k0 = 0; k0 < SP; k0 += 8) {
      v2f a, b;
      a.x = Ash[ll * 177 + k0 + 2 * hi];
      a.y = Ash[ll * 177 + k0 + 2 * hi + 1];
      b.x = Bsh[(k0 + 2 * hi)     * SP + nt16 + ll];
      b.y = Bsh[(k0 + 2 * hi + 1) * SP + nt16 + ll];
      a0 = __builtin_amdgcn_wmma_f32_16x16x4_f32(false, a, false, b, (short)0, a0, false, false);
      a.x = Ash[ll * 177 + k0 + 4 + 2 * hi];
      a.y = Ash[ll * 177 + k0 + 4 + 2 * hi + 1];
      b.x = Bsh[(k0 + 4 + 2 * hi) * SP + nt16 + ll];
      b.y = Bsh[(k0 + 5 + 2 * hi) * SP + nt16 + ll];
      a1 = __builtin_amdgcn_wmma_f32_16x16x4_f32(false, a, false, b, (short)0, a1, false, false);
    }
    const v8f accv = a0 + a1;
    const float* ap = (const float*)&accv;
    const float mk = tv ? msk[t] : 0.f;
#pragma unroll
    for (int r = 0; r < 8; ++r) {
      const int cc = m * 16 + r + 8 * hi;
      const float fvv = Ash[(r + 8 * hi) * 177 + (tv ? t : 0)];
      const float rec = mk * ap[r] + (1.f - mk) * fvv;
      if (tv) {
        if (branch == 0) out[((size_t)n * CCH + cc) * HW + t] = rec;
        const float d = fvv - rec + 1e-6f;
        dacc += d * d;
      }
    }
  }
  dacc += __shfl_xor(dacc, 16, 32);      // combine the two half-wave c-groups (same t)
  if (lane < 16 && tv) ws[WS_DSQ + (branch * NTOT + n) * SP + t] = dacc;
}

// ---------------- Kernel 4a: comask normalization + triplet ----------------
__global__ __launch_bounds__(192) void k_comask(const int* __restrict__ pidx,
                                                float* __restrict__ ws) {
  const int n = blockIdx.x, t = threadIdx.x;
  const int p = pidx[n];
  float raw = 0.f;
  if (t < HW) raw = ws[WS_MW + n * SP + t] * ws[WS_MASK + p * SP + t];
  __shared__ float smin[192], smax[192];
  smin[t] = (t < HW) ? raw : INF;
  smax[t] = (t < HW) ? raw : -INF;
  __syncthreads();
  if (t < 96) { smin[t] = fminf(smin[t], smin[t + 96]); smax[t] = fmaxf(smax[t], smax[t + 96]); }
  __syncthreads();
  for (int s = 48; s > 0; s >>= 1) {
    if (t < s) { smin[t] = fminf(smin[t], smin[t + s]); smax[t] = fmaxf(smax[t], smax[t + s]); }
    __syncthreads();
  }
  const float mn = smin[0], mx = smax[0];
  if (t < HW) {
    ws[WS_CM + n * SP + t] = (raw - mn) / ((mx - mn) + 1e-12f);
    const float dap = sqrtf(ws[WS_DSQ + n * SP + t]);
    const float dan = sqrtf(ws[WS_DSQ + (NTOT + n) * SP + t]);
    ws[WS_TRIP + n * SP + t] = fmaxf(dap - dan + 0.3f, 0.f);
  }
}

// ---------------- Kernel 4b: loss (broadcast semantics: outer product over n) ----------------
__global__ __launch_bounds__(192) void k_loss(const float* __restrict__ ws,
                                              float* __restrict__ out) {
  const int t = threadIdx.x;
  float prod = 0.f;
  if (t < HW) {
    float sc = 0.f, st = 0.f;
    for (int n = 0; n < NTOT; ++n) {
      sc += ws[WS_CM + n * SP + t];
      st += ws[WS_TRIP + n * SP + t];
    }
    prod = sc * st;
  }
  __shared__ float sred[192];
  sred[t] = prod;
  __syncthreads();
  if (t < 96) sred[t] += sred[t + 96];
  __syncthreads();
  for (int s = 48; s > 0; s >>= 1) {
    if (t < s) sred[t] += sred[t + s];
    __syncthreads();
  }
  if (t == 0) out[(size_t)NTOT * CCH * HW] = sred[0] / (float)(NTOT * NTOT * HW);
}

extern "C" void kernel_launch(void* const* d_in, const int* in_sizes, int n_in,
                              void* d_out, int out_size, void* d_ws, size_t ws_size,
                              hipStream_t stream) {
  const float* fv = (const float*)d_in[0];
  const float* ft = (const float*)d_in[1];
  const int* pidx = (const int*)d_in[2];
  const int* nidx = (const int*)d_in[3];
  float* out = (float*)d_out;
  float* ws  = (float*)d_ws;

  const size_t shmem2 = (size_t)(4 * PANEL + 3 * SP) * sizeof(float);          // ~92 KB
  const size_t shmem3 = (size_t)(SP * SP + 16 * 177 + SP) * sizeof(float);     // ~136 KB
  hipFuncSetAttribute((const void*)k_sim,  hipFuncAttributeMaxDynamicSharedMemorySize, (int)shmem2);
  hipFuncSetAttribute((const void*)k_warp, hipFuncAttributeMaxDynamicSharedMemorySize, (int)shmem3);

  k_norm  <<<NTOT,     192, 0,      stream>>>(fv, ft, ws);
  k_sim   <<<2 * NTOT, BLK2, shmem2, stream>>>(fv, ft, pidx, nidx, ws);
  k_warp  <<<2 * NTOT, BLK2, shmem3, stream>>>(fv, ft, ws, out);
  k_comask<<<NTOT,     192, 0,      stream>>>(pidx, ws);
  k_loss  <<<1,        192, 0,      stream>>>(ws, out);
}